// Fuzzy_MultiHeadAttention_QK_69200513073604
// MI455X (gfx1250) — compile-verified
//
#include <hip/hip_runtime.h>
#include <hip/hip_bf16.h>

// B=64, N=1024, D=64, R=64
#define BB 64
#define NN 1024
#define DD 64
#define RR 64

typedef __attribute__((ext_vector_type(2))) float v2f;
typedef __attribute__((ext_vector_type(8))) float v8f;

// -------------------------------------------------------------------------
// Kernel 1: sq_q[b,d] = sum_n query[b,n,d]^2 ; sq_k[r,d] = sum_n keys[r,n,d]^2
// 128 blocks (64 for query, 64 for keys), 256 threads
// -------------------------------------------------------------------------
__global__ void sq_kernel(const float* __restrict__ q, const float* __restrict__ keys,
                          float* __restrict__ sq_q, float* __restrict__ sq_k) {
  const int blk = blockIdx.x;
  const float* src;
  float* dst;
  if (blk < BB) { src = q    + (size_t)blk        * NN * DD; dst = sq_q + blk * DD; }
  else          { src = keys + (size_t)(blk - BB) * NN * DD; dst = sq_k + (blk - BB) * DD; }
  const int d    = threadIdx.x & 63;
  const int part = threadIdx.x >> 6;  // 0..3
  float acc = 0.f;
  for (int n = part; n < NN; n += 4) {
    float v = src[n * DD + d];
    acc += v * v;
  }
  __shared__ float red[256];
  red[threadIdx.x] = acc;
  __syncthreads();
  if (part == 0) dst[d] = red[d] + red[64 + d] + red[128 + d] + red[192 + d];
}

// -------------------------------------------------------------------------
// Kernel 2: cross[b,r,d] = sum_n query[b,n,d]*keys[r,n,d]
// grid 64 = 8 b-tiles x 8 r-tiles; each block: 8b x 8r x 64d outputs
// -------------------------------------------------------------------------
__global__ void cross_kernel(const float* __restrict__ q, const float* __restrict__ keys,
                             float* __restrict__ cross) {
  const int bt = (blockIdx.x >> 3) * 8;  // base b
  const int rt = (blockIdx.x & 7) * 8;   // base r
  __shared__ float qs[8][8][64];  // [b_local][nn][d] 16KB
  __shared__ float ks[8][8][64];  // [r_local][nn][d] 16KB
  const int t  = threadIdx.x;
  const int bl = t >> 5;          // 0..7
  const int rl = (t >> 2) & 7;    // 0..7
  const int dg = (t & 3) * 16;    // d base
  float acc[16];
#pragma unroll
  for (int j = 0; j < 16; j++) acc[j] = 0.f;

  for (int n0 = 0; n0 < NN; n0 += 8) {
    __syncthreads();
#pragma unroll
    for (int i = 0; i < 16; i++) {       // 4096 floats each buffer, 16/thread
      int idx = t + i * 256;
      int lb  = idx >> 9;                // 0..7
      int rem = idx & 511;
      int nn  = rem >> 6;
      int d   = rem & 63;
      qs[lb][nn][d] = q   [(size_t)(bt + lb) * NN * DD + (n0 + nn) * DD + d];
      ks[lb][nn][d] = keys[(size_t)(rt + lb) * NN * DD + (n0 + nn) * DD + d];
    }
    __syncthreads();
#pragma unroll
    for (int nn = 0; nn < 8; nn++) {
#pragma unroll
      for (int j = 0; j < 16; j++)
        acc[j] += qs[bl][nn][dg + j] * ks[rl][nn][dg + j];
    }
  }
#pragma unroll
  for (int j = 0; j < 16; j++)
    cross[((size_t)(bt + bl) * RR + (rt + rl)) * DD + dg + j] = acc[j];
}

// -------------------------------------------------------------------------
// Kernel 3: in-place on cross: attn = softmax_r( -sqrt(max(qq+kk-2*cross,0)) )
// 16 blocks x 256 threads = 4096 (b,d) pairs
// -------------------------------------------------------------------------
__global__ void softmax_kernel(float* __restrict__ cross,
                               const float* __restrict__ sq_q,
                               const float* __restrict__ sq_k) {
  const int idx = blockIdx.x * 256 + threadIdx.x;  // 0..4095
  const int b = idx >> 6, d = idx & 63;
  const float qq = sq_q[b * DD + d];
  float vals[RR];
  float mx = -1e30f;
#pragma unroll 4
  for (int r = 0; r < RR; r++) {
    float c = cross[((size_t)b * RR + r) * DD + d];
    float d2 = fmaxf(qq + sq_k[r * DD + d] - 2.f * c, 0.f);
    float v = -sqrtf(d2);
    vals[r] = v;
    mx = fmaxf(mx, v);
  }
  float s = 0.f;
#pragma unroll 4
  for (int r = 0; r < RR; r++) {
    float e = __expf(vals[r] - mx);
    vals[r] = e;
    s += e;
  }
  const float inv = 1.f / s;
#pragma unroll 4
  for (int r = 0; r < RR; r++)
    cross[((size_t)b * RR + r) * DD + d] = vals[r] * inv;
}

// -------------------------------------------------------------------------
// Kernel 4: per-batch fused matrix
//   M[d][e]  = sum_r attn[b,r,d] * W_v[r*64+d][e]
//   M2[e][f] = sum_d M[d][e] * W_o[f][d]
// one block per batch, 256 threads
// -------------------------------------------------------------------------
__global__ void m2_kernel(const float* __restrict__ attn, const float* __restrict__ W_v,
                          const float* __restrict__ W_o, float* __restrict__ M2) {
  const int b = blockIdx.x;
  __shared__ float a [64][65];   // attn[r][d]
  __shared__ float m [64][65];   // M[d][e]
  __shared__ float wo[64][65];   // W_o[f][d]
  const int t = threadIdx.x;
#pragma unroll
  for (int i = 0; i < 16; i++) {
    int idx = t + i * 256;
    a [idx >> 6][idx & 63] = attn[(size_t)b * 4096 + idx];
    wo[idx >> 6][idx & 63] = W_o[idx];
  }
  __syncthreads();
  {
    const int e = t & 63;
    for (int dd = t >> 6; dd < 64; dd += 4) {
      float acc = 0.f;
#pragma unroll 4
      for (int r = 0; r < 64; r++)
        acc += a[r][dd] * W_v[(size_t)((r << 6) + dd) * 64 + e];
      m[dd][e] = acc;
    }
  }
  __syncthreads();
  {
    const int f = t & 63;
    for (int e = t >> 6; e < 64; e += 4) {
      float acc = 0.f;
#pragma unroll 4
      for (int dd = 0; dd < 64; dd++)
        acc += m[dd][e] * wo[f][dd];
      M2[(size_t)b * 4096 + (e << 6) + f] = acc;
    }
  }
}

// -------------------------------------------------------------------------
// Kernel 5: out[b,n,f] = sum_e value[b,n,e] * M2[b][e][f] + b_o[f]
// WMMA f32 16x16x4. grid = B*4 blocks (256-row segment each), 128 threads
// Per wave: 4 n-tiles x 4 f-tiles, K=64 as 16 chunks of 4.
// A fragments in VGPRs (reused across f-tiles), B fragments from LDS.
// -------------------------------------------------------------------------
__global__ void __launch_bounds__(128)
out_gemm_kernel(const float* __restrict__ value, const float* __restrict__ M2,
                const float* __restrict__ b_o, float* __restrict__ out) {
  const int b   = blockIdx.x >> 2;
  const int seg = blockIdx.x & 3;
  __shared__ float m2[64][64];                       // M2[e][f], 16KB
  const int t = threadIdx.x;
  for (int i = t; i < 4096; i += 128) m2[i >> 6][i & 63] = M2[(size_t)b * 4096 + i];
  __syncthreads();

  const int wave  = t >> 5;
  const int lane  = t & 31;
  const int lhalf = lane >> 4;        // 0|1  -> K pair / row-half select
  const int lm    = lane & 15;        // M (A,C rows) or N (B,C cols) index

  const int nbase0 = seg * 256 + wave * 64;
  for (int nt = 0; nt < 4; ++nt) {
    const int nbase = nbase0 + nt * 16;
    // A 16x4 layout: lane holds A[m=lm][k], VGPR j half h -> k = 2h + j
    float aA[16][2];
    const float* vrow = value + ((size_t)b * NN + nbase + lm) * DD + 2 * lhalf;
#pragma unroll
    for (int kc = 0; kc < 16; ++kc) {
      aA[kc][0] = vrow[kc * 4 + 0];
      aA[kc][1] = vrow[kc * 4 + 1];
    }
#pragma unroll
    for (int ft = 0; ft < 4; ++ft) {
      v8f c = {0.f, 0.f, 0.f, 0.f, 0.f, 0.f, 0.f, 0.f};
      const int fb = ft * 16 + lm;
#pragma unroll
      for (int kc = 0; kc < 16; ++kc) {
        v2f av; av.x = aA[kc][0]; av.y = aA[kc][1];
        // B 4x16 layout: VGPR j half h -> row k = 2h + j, col = lm
        v2f bv;
        bv.x = m2[kc * 4 + 2 * lhalf + 0][fb];
        bv.y = m2[kc * 4 + 2 * lhalf + 1][fb];
        c = __builtin_amdgcn_wmma_f32_16x16x4_f32(
            /*neg_a=*/false, av, /*neg_b=*/false, bv,
            /*c_mod=*/(short)0, c, /*reuse_a=*/false, /*reuse_b=*/false);
      }
      const float bias = b_o[fb];
#pragma unroll
      for (int i = 0; i < 8; i++) {
        const int row = nbase + i + 8 * lhalf;     // C/D: vgpr i -> M=i (+8 for hi half)
        out[((size_t)b * NN + row) * DD + fb] = c[i] + bias;
      }
    }
  }
}

// -------------------------------------------------------------------------
extern "C" void kernel_launch(void* const* d_in, const int* in_sizes, int n_in,
                              void* d_out, int out_size, void* d_ws, size_t ws_size,
                              hipStream_t stream) {
  const float* query = (const float*)d_in[0];
  const float* key   = (const float*)d_in[1];  (void)key;  // unused by reference
  const float* value = (const float*)d_in[2];
  const float* W_v   = (const float*)d_in[3];
  const float* keys  = (const float*)d_in[4];
  const float* W_o   = (const float*)d_in[5];
  const float* b_o   = (const float*)d_in[6];
  float* out = (float*)d_out;
  float* ws  = (float*)d_ws;

  float* sq_q = ws;                      // 4096
  float* sq_k = ws + 4096;               // 4096
  float* attn = ws + 8192;               // 262144 (cross, then softmaxed in place)
  float* M2   = ws + 8192 + 262144;      // 262144

  sq_kernel     <<<128, 256, 0, stream>>>(query, keys, sq_q, sq_k);
  cross_kernel  <<<64,  256, 0, stream>>>(query, keys, attn);
  softmax_kernel<<<16,  256, 0, stream>>>(attn, sq_q, sq_k);
  m2_kernel     <<<64,  256, 0, stream>>>(attn, W_v, W_o, M2);
  out_gemm_kernel<<<BB * 4, 128, 0, stream>>>(value, M2, b_o, out);
}